// CriterionG_40656160424161
// MI455X (gfx1250) — compile-verified
//
#include <hip/hip_runtime.h>
#include <hip/hip_bf16.h>
#include <math.h>

// Problem constants from the reference
#define NCLASS 512
#define EPSV   1e-7f

typedef __attribute__((ext_vector_type(2))) float v2f;
typedef __attribute__((ext_vector_type(8))) float v8f;

// ---------------------------------------------------------------------------
// Kernel 0: zero the workspace accumulators (sums[512] | counts[512]).
// Must run every launch: graph replays do not re-poison d_ws.
// ---------------------------------------------------------------------------
__global__ void crit_zero_ws(float* __restrict__ ws) {
    ws[blockIdx.x * blockDim.x + threadIdx.x] = 0.0f;
}

// ---------------------------------------------------------------------------
// Kernel 1: per-sample gather + nll, block-local segment reduction in LDS
// (ds_add_f32 atomics), then one global f32 atomic per class per block.
// ---------------------------------------------------------------------------
__global__ void crit_main(const float* __restrict__ close_er,
                          const int*   __restrict__ y,
                          const float* __restrict__ max_dis,
                          const float* __restrict__ margin_p,
                          float*       __restrict__ ws,      // [0..511]=sums, [512..1023]=counts
                          int n, int samples_per_thread) {
    __shared__ float lds_sum[NCLASS];
    __shared__ float lds_cnt[NCLASS];
    for (int c = threadIdx.x; c < NCLASS; c += blockDim.x) {
        lds_sum[c] = 0.0f;
        lds_cnt[c] = 0.0f;
    }
    __syncthreads();

    const float margin = margin_p[0];
    const int block_base = (int)(blockIdx.x * blockDim.x) * samples_per_thread;

    // Coalesced: thread t handles block_base + t + k*blockDim.x
    for (int k = 0; k < samples_per_thread; ++k) {
        int nidx = block_base + (int)threadIdx.x + k * (int)blockDim.x;
        if (nidx < n) {
            int   cls = y[nidx];
            float ce  = close_er[(size_t)nidx * NCLASS + (size_t)cls];
            float md  = max_dis[cls];
            float x   = ce - md - margin;
            // sigmoid; saturates cleanly (exp overflow -> inf -> g=0 -> clamped)
            float g   = 1.0f / (1.0f + expf(-x));
            g = fminf(fmaxf(g, EPSV), 1.0f - EPSV);
            float nll = -logf(g);
            atomicAdd(&lds_sum[cls], nll);   // ds_add_f32
            atomicAdd(&lds_cnt[cls], 1.0f);  // ds_add_f32
        }
    }
    __syncthreads();

    for (int c = threadIdx.x; c < NCLASS; c += blockDim.x) {
        atomicAdd(&ws[c],          lds_sum[c]);  // global_atomic_add_f32
        atomicAdd(&ws[NCLASS + c], lds_cnt[c]);
    }
}

// ---------------------------------------------------------------------------
// Kernel 2: finalize with a WMMA F32 16x16x4 reduction chain (single wave32).
//
// With B = ones(4x16):  D[m,n] = sum_k A[m,k] + C[m,n]  -- i.e. the matrix
// unit performs a 4-way K reduction per row, accumulating in f32. We chain
// 16 WMMAs: per chunk i, A-rows 0..7 carry 32 per-class means and A-rows
// 8..15 carry the 32 matching non-empty flags. After the chain:
//   sum over D rows 0..7  = sum of all 512 means   (lane 0, VGPRs 0..7, N=0)
//   sum over D rows 8..15 = j (# non-empty classes) (lane 16, VGPRs 0..7, N=0)
//
// A-matrix 16x4 f32 layout (ISA 7.12.2): lane l<16 holds {K=0,K=1} of row M=l
// in VGPR0/VGPR1; lane l+16 holds {K=2,K=3} of row M=l.
// EXEC is all-ones: 32 threads, no divergent branches (ternaries -> cndmask).
// ---------------------------------------------------------------------------
__global__ void crit_finalize(const float* __restrict__ ws, float* __restrict__ out) {
    const int  lane = (int)threadIdx.x;   // 0..31
    const int  l    = lane & 15;
    const int  koff = (lane >= 16) ? 2 : 0;
    const bool flag_row = (l >= 8);

    v2f ones;
    ones.x = 1.0f; ones.y = 1.0f;

    v8f acc = {};
    #pragma unroll
    for (int i = 0; i < 16; ++i) {
        // This lane feeds A[M, K=koff..koff+1]; its class pair:
        int cls = i * 32 + (l & 7) * 4 + koff;

        float s0 = ws[cls];
        float c0 = ws[NCLASS + cls];
        float s1 = ws[cls + 1];
        float c1 = ws[NCLASS + cls + 1];

        float mean0 = (c0 > 0.0f) ? (s0 / c0) : 0.0f;
        float mean1 = (c1 > 0.0f) ? (s1 / c1) : 0.0f;
        float f0    = (c0 > 0.0f) ? 1.0f : 0.0f;
        float f1    = (c1 > 0.0f) ? 1.0f : 0.0f;

        v2f a;
        a.x = flag_row ? f0 : mean0;
        a.y = flag_row ? f1 : mean1;

        // 8 args: (neg_a, A, neg_b, B, c_mod, C, reuse_a, reuse_b)
        acc = __builtin_amdgcn_wmma_f32_16x16x4_f32(
            false, a, false, ones, (short)0, acc, false, false);
    }

    // Per-lane: sum this lane's 8 accumulator elements (N = l column).
    float s = acc[0] + acc[1] + acc[2] + acc[3] + acc[4] + acc[5] + acc[6] + acc[7];

    float sum_means = __shfl(s, 0, 32);   // rows 0..7  @ N=0
    float j         = __shfl(s, 16, 32);  // rows 8..15 @ N=0

    if (lane == 0) out[0] = sum_means / j;
}

// ---------------------------------------------------------------------------
extern "C" void kernel_launch(void* const* d_in, const int* in_sizes, int n_in,
                              void* d_out, int out_size, void* d_ws, size_t ws_size,
                              hipStream_t stream) {
    const float* close_er = (const float*)d_in[0];
    const int*   y        = (const int*)  d_in[1];
    const float* max_dis  = (const float*)d_in[2];
    const float* margin   = (const float*)d_in[3];
    float*       out      = (float*)d_out;
    float*       ws       = (float*)d_ws;   // needs 1024 floats = 4 KB

    const int n = in_sizes[1];              // number of samples (y is [N])

    // Zero sums/counts
    crit_zero_ws<<<dim3(4), dim3(256), 0, stream>>>(ws);

    // Main pass: 256 threads (8 wave32), 8 samples/thread
    const int TPB = 256;
    const int SPT = 8;
    const int blocks = (n + TPB * SPT - 1) / (TPB * SPT);
    crit_main<<<dim3(blocks), dim3(TPB), 0, stream>>>(
        close_er, y, max_dis, margin, ws, n, SPT);

    // Finalize: single wave32 using the WMMA f32 reduction chain
    crit_finalize<<<dim3(1), dim3(32), 0, stream>>>(ws, out);
}